// CTCLoss_49280454754906
// MI455X (gfx1250) — compile-verified
//
#include <hip/hip_runtime.h>
#include <stdint.h>

// Problem shape (fixed by the reference).
#define T_DIM 1024
#define B_DIM 128
#define C_DIM 128
#define L_DIM 512
#define NTH   256               // 8 wave32s per workgroup
#define BLANK (C_DIM - 1)
#define NEGF  (-1.0e30f)

// ---- CDNA5 async global->LDS copy (ASYNCcnt path) -------------------------
// GVS mode: addr = SGPR64 base + VGPR32 byte offset. VDST holds the LDS byte
// address (low 32 bits of the flat shared-aperture address).
__device__ __forceinline__ void async_row_b128(const float* saddr,
                                               uint32_t voff_bytes,
                                               const float* lds_dst) {
  uint32_t lds32 = (uint32_t)(uintptr_t)lds_dst;
  asm volatile("global_load_async_to_lds_b128 %0, %1, %2"
               :
               : "v"(lds32), "v"(voff_bytes), "s"(saddr)
               : "memory");
}
// Async loads complete in order -> waiting cnt<=N with N+1 outstanding
// guarantees the oldest landed.
__device__ __forceinline__ void wait_async_le0() { asm volatile("s_wait_asynccnt 0" ::: "memory"); }
__device__ __forceinline__ void wait_async_le1() { asm volatile("s_wait_asynccnt 1" ::: "memory"); }
__device__ __forceinline__ void wait_async_le2() { asm volatile("s_wait_asynccnt 2" ::: "memory"); }

// 3-way logaddexp (mirrors the reference's nested logaddexp; NEG sentinel
// underflows to exp()==0 exactly as in the -1e30 log-space reference).
__device__ __forceinline__ float la3(float a, float b, float c) {
  float m = fmaxf(fmaxf(a, b), c);
  return m + __logf(__expf(a - m) + __expf(b - m) + __expf(c - m));
}

// Wave32 logsumexp of 128 floats held as float4 per lane.
__device__ __forceinline__ float wave_lse4(float4 v) {
  float m = fmaxf(fmaxf(v.x, v.y), fmaxf(v.z, v.w));
  for (int off = 16; off > 0; off >>= 1) m = fmaxf(m, __shfl_xor(m, off, 32));
  float s = __expf(v.x - m) + __expf(v.y - m) + __expf(v.z - m) + __expf(v.w - m);
  for (int off = 16; off > 0; off >>= 1) s += __shfl_xor(s, off, 32);
  return m + __logf(s);
}

__global__ __launch_bounds__(NTH) void ctc_alpha_kernel(
    const float* __restrict__ y_pred,   // [T, B, C] fp32
    const int*   __restrict__ y_true,   // [L, B] int32
    float*       __restrict__ ll_out) { // [B]
  __shared__ __align__(16) float s_row[4][C_DIM];   // 4-slot emission-row ring
  __shared__ float s_bnd[2][2 * NTH + 2];           // boundary alpha pairs
  __shared__ float s_lse[2];                        // pipelined logsumexp

  const int tid = threadIdx.x;
  const int b   = blockIdx.x;

  // ---- per-thread state metadata: classes + allow2 mask -------------------
  // Thread tid owns states 4*tid..4*tid+3; thread 255 additionally owns 1024.
  const int sBase = tid * 4;
  const int n = (tid == NTH - 1) ? 5 : 4;
  int cls[5];
  unsigned allow2 = 0u;
#pragma unroll
  for (int i = 0; i < 5; ++i) {
    int s = sBase + i;
    int c = BLANK;
    if (i < n && (s & 1)) {
      int li = (s - 1) >> 1;
      c = y_true[li * B_DIM + b];
      int p2 = (s >= 3) ? y_true[(li - 1) * B_DIM + b] : BLANK;
      if (c != BLANK && c != p2) allow2 |= (1u << i);
    }
    cls[i] = c;
  }

  if (tid == 0) {  // states -2,-1 for thread 0: always NEG (never rewritten)
    s_bnd[0][0] = NEGF; s_bnd[0][1] = NEGF;
    s_bnd[1][0] = NEGF; s_bnd[1][1] = NEGF;
  }

  // ---- prologue: issue rows 0..2, lse for rows 0 and 1, issue row 3 -------
  if (tid < 32) {
    const uint32_t voff = (uint32_t)(tid * 16);
    async_row_b128(y_pred + ((size_t)0 * B_DIM + b) * C_DIM, voff, &s_row[0][tid * 4]);
    async_row_b128(y_pred + ((size_t)1 * B_DIM + b) * C_DIM, voff, &s_row[1][tid * 4]);
    async_row_b128(y_pred + ((size_t)2 * B_DIM + b) * C_DIM, voff, &s_row[2][tid * 4]);
    wait_async_le2();  // row 0 landed
    float lse0 = wave_lse4(*(const float4*)&s_row[0][tid * 4]);
    wait_async_le1();  // row 1 landed
    float lse1 = wave_lse4(*(const float4*)&s_row[1][tid * 4]);
    if (tid == 0) { s_lse[0] = lse0; s_lse[1] = lse1; }
    async_row_b128(y_pred + ((size_t)3 * B_DIM + b) * C_DIM, voff, &s_row[3][tid * 4]);
  }
  __syncthreads();

  // ---- alpha init (t = 0): only states 0 and 1 are live -------------------
  float a[5];
  {
    const float lse0 = s_lse[0];
#pragma unroll
    for (int i = 0; i < 5; ++i) {
      int s = sBase + i;
      float av = NEGF;
      if (i < n && s < 2) av = s_row[0][cls[i]] - lse0;
      a[i] = av;
    }
  }
  s_bnd[0][2 * tid + 2] = a[2];  // alpha[4*tid+2], alpha[4*tid+3] for tid+1
  s_bnd[0][2 * tid + 3] = a[3];
  __syncthreads();

  // ---- sequential scan: ONE barrier per step ------------------------------
  // Step t: gathers read s_row[t&3] with s_lse[t&1] (published at end of
  // step t-1). Wave 0 concurrently computes lse_{t+1} from s_row[(t+1)&3]
  // and issues row t+3 into s_row[(t+3)&3] (== row t-1's slot, fully
  // consumed last step). All four slots are distinct mod 4.
  for (int t = 1; t < T_DIM; ++t) {
    const float  lse  = s_lse[t & 1];
    const float* row  = s_row[t & 3];
    const float* bndp = s_bnd[(t - 1) & 1];
    float pm2 = bndp[2 * tid];       // alpha_{t-1}[sBase-2]
    float pm1 = bndp[2 * tid + 1];   // alpha_{t-1}[sBase-1]
#pragma unroll
    for (int i = 0; i < 5; ++i) {
      if (i < n) {
        float lp  = row[cls[i]] - lse;                  // emission gather
        float s2v = (allow2 & (1u << i)) ? pm2 : NEGF;  // skip transition
        float av  = a[i];
        float r   = la3(av, pm1, s2v);
        pm2 = pm1; pm1 = av;                            // shift old alphas
        a[i] = lp + r;
      }
    }
    s_bnd[t & 1][2 * tid + 2] = a[2];
    s_bnd[t & 1][2 * tid + 3] = a[3];

    if (tid < 32 && t + 1 < T_DIM) {     // pipeline lse one step ahead
      if (t + 2 < T_DIM) wait_async_le1(); else wait_async_le0();
      float nl = wave_lse4(*(const float4*)&s_row[(t + 1) & 3][tid * 4]);
      if (tid == 0) s_lse[(t + 1) & 1] = nl;
      if (t + 3 < T_DIM) {
        async_row_b128(y_pred + ((size_t)(t + 3) * B_DIM + b) * C_DIM,
                       (uint32_t)(tid * 16), &s_row[(t + 3) & 3][tid * 4]);
      }
    }
    __syncthreads();  // publishes bnd[t&1], s_lse[(t+1)&1], row t+1 data
  }

  if (tid == NTH - 1) {  // a[4]=alpha_T[1024], a[3]=alpha_T[1023]
    float x = a[4], y = a[3];
    float m = fmaxf(x, y);
    ll_out[b] = m + __logf(__expf(x - m) + __expf(y - m));
  }
}

// Deterministic mean over the 128 per-batch log-likelihoods.
__global__ __launch_bounds__(B_DIM) void ctc_reduce_kernel(
    const float* __restrict__ ll, float* __restrict__ out) {
  __shared__ float sm[B_DIM];
  const int tid = threadIdx.x;
  sm[tid] = ll[tid];
  __syncthreads();
  for (int off = B_DIM / 2; off > 0; off >>= 1) {
    if (tid < off) sm[tid] += sm[tid + off];
    __syncthreads();
  }
  if (tid == 0) out[0] = -sm[0] / (float)B_DIM;
}

extern "C" void kernel_launch(void* const* d_in, const int* in_sizes, int n_in,
                              void* d_out, int out_size, void* d_ws, size_t ws_size,
                              hipStream_t stream) {
  (void)in_sizes; (void)n_in; (void)out_size; (void)ws_size;
  const float* y_pred = (const float*)d_in[0];
  const int*   y_true = (const int*)d_in[1];
  float* ll = (float*)d_ws;  // 128 floats of scratch

  ctc_alpha_kernel<<<B_DIM, NTH, 0, stream>>>(y_pred, y_true, ll);
  ctc_reduce_kernel<<<1, B_DIM, 0, stream>>>(ll, (float*)d_out);
}